// Match_Decoder_48876727829078
// MI455X (gfx1250) — compile-verified
//
#include <hip/hip_runtime.h>
#include <math.h>

typedef __attribute__((ext_vector_type(2))) float v2f;
typedef __attribute__((ext_vector_type(8))) float v8f;
typedef unsigned int u32x4 __attribute__((ext_vector_type(4)));
typedef int i32x8 __attribute__((ext_vector_type(8)));
typedef int i32x4 __attribute__((ext_vector_type(4)));

#define H 1024
#define SRC 4096
#define VOCAB 32000

#define TILE_ROWS 128     // rows per block (8 waves x 16 rows)
#define CHUNK_K   32      // K chunk staged in LDS per iteration
#define LDS_STRIDE 36     // 32 + 4 pad floats (144B rows, 16B aligned, no bank conflicts)

#if defined(__has_builtin)
#if __has_builtin(__builtin_amdgcn_tensor_load_to_lds)
#define USE_TDM 1
#endif
#endif

#ifdef USE_TDM
// ---------------------------------------------------------------------------
// Issue one TDM descriptor: load a TILE_ROWS x CHUNK_K f32 tile from a
// row-major matrix (row stride K floats) into LDS at lds_off, with hardware
// padding of 4 DWORDs every 32 DWORDs (-> LDS row stride 36 floats).
// D# packing per ISA 8.3/8.4. Groups 2/3 zero => 2D tile. Issued by one wave;
// completion tracked with TENSORcnt. (clang-23 6-arg builtin form.)
// ---------------------------------------------------------------------------
__device__ __forceinline__ void tdm_issue_tile(const float* gsrc, unsigned lds_off, int K)
{
  unsigned long long ga = (unsigned long long)(uintptr_t)gsrc;
  u32x4 g0;
  g0[0] = 1u;                                   // count=1 (valid), no gather
  g0[1] = lds_off;                              // lds_addr (bytes)
  g0[2] = (unsigned)ga;                         // global_addr[31:0]
  g0[3] = (unsigned)((ga >> 32) & 0x1FFFFFFull) // global_addr[56:32]
          | 0x80000000u;                        // type=2 ("image") in bits[127:126]
  i32x8 g1;
  g1[0] = (int)((2u << 16)       // data_size = 4 bytes
              | (1u << 20)       // pad_enable
              | (4u << 22)       // pad_interval code 4 => every 32 DWORDs
              | (3u << 25));     // pad_amount  code 3 => 4 DWORDs
  g1[1] = (int)((unsigned)CHUNK_K << 16);   // tensor_dim0[15:0] = 32
  g1[2] = (int)((unsigned)TILE_ROWS << 16); // tensor_dim0 hi=0 | tensor_dim1[15:0]=128
  g1[3] = (int)((unsigned)CHUNK_K << 16);   // tensor_dim1 hi=0 | tile_dim0=32
  g1[4] = (int)TILE_ROWS;                   // tile_dim1=128 | tile_dim2=0
  g1[5] = K;                                // tensor_dim0_stride[31:0] (floats)
  g1[6] = 0;                                // stride hi | tensor_dim1_stride lo
  g1[7] = 0;
  i32x4 gz4 = {0, 0, 0, 0};
  i32x8 gz8 = {0, 0, 0, 0, 0, 0, 0, 0};
  __builtin_amdgcn_tensor_load_to_lds(g0, g1, gz4, gz4, gz8, 0);
}
#endif

// ---------------------------------------------------------------------------
// Generic GEMV y[r] = dot(W[r,:], x) (+ bias) using V_WMMA_F32_16X16X4_F32.
// 256 threads = 8 waves, 128 rows/block. W tiles staged into LDS — by the
// Tensor Data Mover (double-buffered, overlapped with WMMA) when available,
// else by coalesced float4 loads. A follows the ISA 16x4 f32 layout
// (lane%16=M, lane/16=K-half, v0/v1 pack K within half); B = x replicated
// across all 16 N columns with the same symmetric K slotting, so every column
// of D equals the GEMV result; lanes 0/16 extract M=v / M=v+8.
// Requires rows % 128 == 0, K % 32 == 0, K <= 2048. EXEC all-ones for WMMA.
// ---------------------------------------------------------------------------
__global__ __launch_bounds__(256) void gemv_wmma_kernel(
    const float* __restrict__ W, const float* __restrict__ x,
    const float* __restrict__ bias, float* __restrict__ y, int K)
{
#ifdef USE_TDM
  __shared__ float sW[2][TILE_ROWS * LDS_STRIDE];
#else
  __shared__ float sW[1][TILE_ROWS * LDS_STRIDE];
#endif
  __shared__ float sX[2048];

  const int tid  = threadIdx.x;
  const int wave = tid >> 5;
  const int lane = tid & 31;
  const int m    = lane & 15;
  const int half = lane >> 4;
  const int rowBase = blockIdx.x * TILE_ROWS;
  const int nchunks = K / CHUNK_K;

  for (int i = tid; i < K; i += 256) sX[i] = x[i];

  v8f acc = {};  // C starts at zero

#ifdef USE_TDM
  const unsigned lds_base = (unsigned)(uintptr_t)&sW[0][0]; // low 32 bits = LDS offset
  const unsigned buf_bytes = TILE_ROWS * LDS_STRIDE * 4u;
  const float* gbase = W + (size_t)rowBase * K;

  if (tid == 0) {                      // wave 0 drives the DMA engine
    tdm_issue_tile(gbase, lds_base, K);
    __builtin_amdgcn_s_wait_tensorcnt(0);
  }
  __syncthreads();

  int buf = 0;
  for (int c = 0; c < nchunks; ++c) {
    if (tid == 0 && c + 1 < nchunks)   // prefetch next chunk into other buffer
      tdm_issue_tile(gbase + (size_t)(c + 1) * CHUNK_K, lds_base + (buf ^ 1) * buf_bytes, K);

    const float* wrow = &sW[buf][(wave * 16 + m) * LDS_STRIDE + 2 * half];
    const float* xrow = &sX[c * CHUNK_K + 2 * half];
    #pragma unroll
    for (int kk = 0; kk < CHUNK_K; kk += 4) {
      v2f a = *(const v2f*)(wrow + kk);
      v2f b = *(const v2f*)(xrow + kk);
      acc = __builtin_amdgcn_wmma_f32_16x16x4_f32(
          false, a, false, b, (short)0, acc, false, false);
    }
    __syncthreads();                               // all done reading buf
    if (tid == 0 && c + 1 < nchunks)
      __builtin_amdgcn_s_wait_tensorcnt(0);        // next tile landed
    __syncthreads();
    buf ^= 1;
  }
#else
  for (int c = 0; c < nchunks; ++c) {
    __syncthreads();
    #pragma unroll
    for (int i = 0; i < (TILE_ROWS * CHUNK_K / 4) / 256; ++i) {
      int idx = tid + i * 256;
      int r   = idx >> 3;               // 8 float4 per 32-float row
      int c4  = idx & 7;
      const float4 v = *(const float4*)(W + (size_t)(rowBase + r) * K + c * CHUNK_K + c4 * 4);
      *(float4*)&sW[0][r * LDS_STRIDE + c4 * 4] = v;
    }
    __syncthreads();
    const float* wrow = &sW[0][(wave * 16 + m) * LDS_STRIDE + 2 * half];
    const float* xrow = &sX[c * CHUNK_K + 2 * half];
    #pragma unroll
    for (int kk = 0; kk < CHUNK_K; kk += 4) {
      v2f a = *(const v2f*)(wrow + kk);
      v2f b = *(const v2f*)(xrow + kk);
      acc = __builtin_amdgcn_wmma_f32_16x16x4_f32(
          false, a, false, b, (short)0, acc, false, false);
    }
  }
#endif

  // D layout: VGPR v, lanes 0-15 -> M=v (all N equal), lanes 16-31 -> M=v+8.
  if (lane == 0 || lane == 16) {
    int r0 = rowBase + wave * 16 + half * 8;
    #pragma unroll
    for (int v = 0; v < 8; ++v) {
      float val = acc[v];
      if (bias) val += bias[r0 + v];
      y[r0 + v] = val;
    }
  }
}

// ---------------------------------------------------------------------------
// rnn_in = concat(emb[word], last_context)
// ---------------------------------------------------------------------------
__global__ __launch_bounds__(256) void prep_kernel(
    const int* __restrict__ word, const float* __restrict__ emb,
    const float* __restrict__ lastctx, float* __restrict__ rnn_in)
{
  const int w = word[0];
  #pragma unroll
  for (int i = 0; i < 8; ++i) {
    int j = threadIdx.x + i * 256;
    rnn_in[j] = (j < H) ? emb[(size_t)w * H + j] : lastctx[j - H];
  }
}

// ---------------------------------------------------------------------------
// GRU gates: h_new -> joined[0:1024] and d_out hidden slot
// ---------------------------------------------------------------------------
__global__ __launch_bounds__(256) void gates_kernel(
    const float* __restrict__ gi, const float* __restrict__ gh,
    const float* __restrict__ hprev, float* __restrict__ joined,
    float* __restrict__ out_hidden)
{
  #pragma unroll
  for (int i = 0; i < 4; ++i) {
    int j = threadIdx.x + i * 256;
    float r = 1.0f / (1.0f + expf(-(gi[j] + gh[j])));
    float z = 1.0f / (1.0f + expf(-(gi[H + j] + gh[H + j])));
    float n = tanhf(gi[2 * H + j] + r * gh[2 * H + j]);
    float h = (1.0f - z) * n + z * hprev[j];
    joined[j] = h;
    out_hidden[j] = h;
  }
}

// ---------------------------------------------------------------------------
// softmax over 4096 scores -> weights (ws) and attn_weights output
// ---------------------------------------------------------------------------
__global__ __launch_bounds__(1024) void softmax4096_kernel(
    const float* __restrict__ s, float* __restrict__ w, float* __restrict__ out_w)
{
  __shared__ float red[1024];
  const int t = threadIdx.x;
  float v0 = s[t], v1 = s[t + 1024], v2 = s[t + 2048], v3 = s[t + 3072];
  float mx = fmaxf(fmaxf(v0, v1), fmaxf(v2, v3));
  red[t] = mx; __syncthreads();
  for (int off = 512; off > 0; off >>= 1) {
    if (t < off) red[t] = fmaxf(red[t], red[t + off]);
    __syncthreads();
  }
  mx = red[0]; __syncthreads();
  float e0 = expf(v0 - mx), e1 = expf(v1 - mx), e2 = expf(v2 - mx), e3 = expf(v3 - mx);
  red[t] = e0 + e1 + e2 + e3; __syncthreads();
  for (int off = 512; off > 0; off >>= 1) {
    if (t < off) red[t] += red[t + off];
    __syncthreads();
  }
  float inv = 1.0f / red[0];
  w[t] = e0 * inv;          out_w[t] = e0 * inv;
  w[t + 1024] = e1 * inv;   out_w[t + 1024] = e1 * inv;
  w[t + 2048] = e2 * inv;   out_w[t + 2048] = e2 * inv;
  w[t + 3072] = e3 * inv;   out_w[t + 3072] = e3 * inv;
}

// ---------------------------------------------------------------------------
// attn context: partials over row chunks (deterministic two-phase reduction)
// ---------------------------------------------------------------------------
__global__ __launch_bounds__(256) void ctx_partial_kernel(
    const float* __restrict__ enc, const float* __restrict__ w,
    float* __restrict__ part)
{
  const int b = blockIdx.x;
  const int t = threadIdx.x;
  float acc0 = 0.f, acc1 = 0.f, acc2 = 0.f, acc3 = 0.f;
  for (int r = b * 128; r < b * 128 + 128; ++r) {
    float wr = w[r];
    const float* row = enc + (size_t)r * H;
    acc0 += wr * row[t];
    acc1 += wr * row[t + 256];
    acc2 += wr * row[t + 512];
    acc3 += wr * row[t + 768];
  }
  float* p = part + (size_t)b * H;
  p[t] = acc0; p[t + 256] = acc1; p[t + 512] = acc2; p[t + 768] = acc3;
}

__global__ __launch_bounds__(256) void ctx_reduce_kernel(
    const float* __restrict__ part, float* __restrict__ joined,
    float* __restrict__ out_ctx)
{
  const int t = threadIdx.x;
  #pragma unroll
  for (int j = 0; j < 4; ++j) {
    int c = t + j * 256;
    float s = 0.f;
    for (int b = 0; b < 32; ++b) s += part[(size_t)b * H + c];
    joined[H + c] = s;
    out_ctx[c] = s;
  }
}

// ---------------------------------------------------------------------------
// in-place log_softmax over 32000 logits (fixed-order -> deterministic)
// ---------------------------------------------------------------------------
__global__ __launch_bounds__(1024) void logsoftmax_kernel(float* __restrict__ logits)
{
  __shared__ float red[1024];
  const int t = threadIdx.x;
  float mx = -INFINITY;
  for (int i = t; i < VOCAB; i += 1024) mx = fmaxf(mx, logits[i]);
  red[t] = mx; __syncthreads();
  for (int off = 512; off > 0; off >>= 1) {
    if (t < off) red[t] = fmaxf(red[t], red[t + off]);
    __syncthreads();
  }
  mx = red[0]; __syncthreads();
  float s = 0.f;
  for (int i = t; i < VOCAB; i += 1024) s += expf(logits[i] - mx);
  red[t] = s; __syncthreads();
  for (int off = 512; off > 0; off >>= 1) {
    if (t < off) red[t] += red[t + off];
    __syncthreads();
  }
  float lse = mx + logf(red[0]);
  for (int i = t; i < VOCAB; i += 1024) logits[i] -= lse;
}

// ---------------------------------------------------------------------------
extern "C" void kernel_launch(void* const* d_in, const int* in_sizes, int n_in,
                              void* d_out, int out_size, void* d_ws, size_t ws_size,
                              hipStream_t stream)
{
  (void)in_sizes; (void)n_in; (void)out_size; (void)ws_size;
  const int*   word    = (const int*)  d_in[0];
  const float* lastctx = (const float*)d_in[1];
  const float* hprev   = (const float*)d_in[2];
  const float* enc     = (const float*)d_in[3];
  const float* emb     = (const float*)d_in[4];
  const float* W_ih    = (const float*)d_in[5];
  const float* W_hh    = (const float*)d_in[6];
  const float* b_ih    = (const float*)d_in[7];
  const float* b_hh    = (const float*)d_in[8];
  const float* W_out   = (const float*)d_in[9];
  const float* b_out   = (const float*)d_in[10];

  float* out = (float*)d_out;
  float* ws  = (float*)d_ws;

  // workspace layout (floats)
  float* rnn_in  = ws + 0;       // 2048
  float* gi      = ws + 2048;    // 3072
  float* gh      = ws + 5120;    // 3072
  float* joined  = ws + 8192;    // 2048 ([h_new, context])
  float* scores  = ws + 10240;   // 4096
  float* weights = ws + 14336;   // 4096
  float* part    = ws + 18432;   // 32*1024

  // output layout: [logp 32000][attn_context 1024][hidden 1024][attn_weights 4096]
  float* out_logp = out;
  float* out_ctx  = out + VOCAB;
  float* out_hid  = out + VOCAB + H;
  float* out_attw = out + VOCAB + 2 * H;

  prep_kernel<<<1, 256, 0, stream>>>(word, emb, lastctx, rnn_in);
  gemv_wmma_kernel<<<3072 / TILE_ROWS, 256, 0, stream>>>(W_ih, rnn_in, b_ih, gi, 2048);
  gemv_wmma_kernel<<<3072 / TILE_ROWS, 256, 0, stream>>>(W_hh, hprev, b_hh, gh, 1024);
  gates_kernel<<<1, 256, 0, stream>>>(gi, gh, hprev, joined, out_hid);
  gemv_wmma_kernel<<<SRC / TILE_ROWS, 256, 0, stream>>>(enc, joined, nullptr, scores, 1024);
  softmax4096_kernel<<<1, 1024, 0, stream>>>(scores, weights, out_attw);
  ctx_partial_kernel<<<32, 256, 0, stream>>>(enc, weights, part);
  ctx_reduce_kernel<<<1, 256, 0, stream>>>(part, joined, out_ctx);
  gemv_wmma_kernel<<<VOCAB / TILE_ROWS, 256, 0, stream>>>(W_out, joined, b_out, out_logp, 2048);
  logsoftmax_kernel<<<1, 1024, 0, stream>>>(out_logp);
}